// Context_23897198034953
// MI455X (gfx1250) — compile-verified
//
#include <hip/hip_runtime.h>
#include <hip/hip_bf16.h>

#define BATCH 2048
#define DIM   256
#define UNITS 256
#define NCTX  64
#define MAXT  192   // sum_c ceil(n_c/16) <= 2048/16 + 63 = 191

typedef float v2f __attribute__((ext_vector_type(2)));
typedef float v8f __attribute__((ext_vector_type(8)));

struct TileDesc { int ctx; int start; int rows; int pad; };

// ---------------------------------------------------------------------------
// Pass 1: bucket samples by context, build M-tile descriptors.
// Single block, 256 threads. Output order within a context is nondeterministic
// but each sample's computed row is independent of its tile position, so the
// final output is deterministic.
// ---------------------------------------------------------------------------
__global__ void __launch_bounds__(256)
bucket_kernel(const int* __restrict__ cidx,
              int* __restrict__ order,
              TileDesc* __restrict__ tiles,
              int* __restrict__ num_tiles) {
    __shared__ int cnt[NCTX];
    __shared__ int cur[NCTX];
    const int t = threadIdx.x;

    if (t < NCTX) cnt[t] = 0;
    __syncthreads();

    for (int i = t; i < BATCH; i += 256)
        atomicAdd(&cnt[cidx[i]], 1);
    __syncthreads();

    if (t == 0) {
        int run = 0, nt = 0;
        for (int c = 0; c < NCTX; ++c) {
            cur[c] = run;
            const int n = cnt[c];
            for (int s = 0; s < n; s += 16) {
                tiles[nt].ctx   = c;
                tiles[nt].start = run + s;
                tiles[nt].rows  = (n - s < 16) ? (n - s) : 16;
                tiles[nt].pad   = 0;
                ++nt;
            }
            run += n;
        }
        *num_tiles = nt;
    }
    __syncthreads();

    for (int i = t; i < BATCH; i += 256) {
        const int c = cidx[i];
        const int p = atomicAdd(&cur[c], 1);
        order[p] = i;
    }
}

// ---------------------------------------------------------------------------
// Pass 2: per-tile GEMM. Block = 256 threads = 8 waves.
// Block computes rows [start, start+rows) of one context, all 256 columns.
// Wave w owns columns [32w, 32w+32) as two 16x16 WMMA accumulators.
// K-loop: 64 steps of V_WMMA_F32_16X16X4_F32 (fp32 in, fp32 acc).
// ---------------------------------------------------------------------------
__global__ void __launch_bounds__(256)
ctx_gemm_kernel(const float* __restrict__ x,
                const float* __restrict__ kern,
                const int* __restrict__ order,
                const TileDesc* __restrict__ tiles,
                const int* __restrict__ num_tiles,
                float* __restrict__ out) {
    __shared__ float As[16][DIM + 4];   // +4 floats row pad: kills LDS bank conflicts
    __shared__ int   samp[16];

    const int tile_id = blockIdx.x;
    if (tile_id >= *num_tiles) return;          // block-uniform exit, EXEC stays full
    const TileDesc td = tiles[tile_id];

    const int t = threadIdx.x;
    if (t < 16) samp[t] = (t < td.rows) ? order[td.start + t] : -1;
    __syncthreads();

    // Stage 16 gathered x-rows into LDS (zero-fill past-end rows).
    // 16 rows x 64 float4 = 1024 float4; 4 per thread, fully coalesced per row.
    #pragma unroll
    for (int j = 0; j < 4; ++j) {
        const int i  = t + 256 * j;
        const int r  = i >> 6;
        const int c4 = (i & 63) << 2;
        float4 v = make_float4(0.f, 0.f, 0.f, 0.f);
        const int s = samp[r];
        if (s >= 0) v = *(const float4*)(x + (size_t)s * DIM + c4);
        *(float4*)(&As[r][c4]) = v;
    }
    __syncthreads();

    const int wave = t >> 5;
    const int lane = t & 31;
    const int half = lane >> 4;     // 0: K pair {0,1}, 1: K pair {2,3}
    const int lid  = lane & 15;     // A row (M) index / B column index
    const int n0   = wave * 32;

    const float* __restrict__ W = kern + (size_t)td.ctx * DIM * UNITS;

    v8f c0 = {};
    v8f c1 = {};

    #pragma unroll 4
    for (int kk = 0; kk < DIM / 4; ++kk) {
        const int krow = kk * 4 + 2 * half;

        // A 16x4: lane holds row `lid`, K = {krow, krow+1} -> ds_load_b64
        v2f a = *(const v2f*)(&As[lid][krow]);

        // B 4x16: lane holds col n0+lid (and +16), K = {krow, krow+1}
        const float* wp = W + (size_t)krow * UNITS + n0 + lid;
        v2f b0, b1;
        b0.x = wp[0];          b0.y = wp[UNITS];
        b1.x = wp[16];         b1.y = wp[UNITS + 16];

        c0 = __builtin_amdgcn_wmma_f32_16x16x4_f32(false, a, false, b0,
                                                   (short)0, c0, false, false);
        c1 = __builtin_amdgcn_wmma_f32_16x16x4_f32(false, a, false, b1,
                                                   (short)0, c1, false, false);
    }

    // Scatter: C/D VGPR v holds row (v + 8*half), col n0 + lid (and +16).
    #pragma unroll
    for (int v = 0; v < 8; ++v) {
        const int r = v + 8 * half;
        if (r < td.rows) {
            const size_t o = (size_t)samp[r] * UNITS + n0 + lid;
            out[o]      = c0[v];
            out[o + 16] = c1[v];
        }
    }
}

extern "C" void kernel_launch(void* const* d_in, const int* in_sizes, int n_in,
                              void* d_out, int out_size, void* d_ws, size_t ws_size,
                              hipStream_t stream) {
    const int*   cidx = (const int*)d_in[0];     // content_idx (2048,1) int32
    const float* x    = (const float*)d_in[1];   // (2048, 256) f32
    const float* kern = (const float*)d_in[2];   // (64, 256, 256) f32
    float*       out  = (float*)d_out;           // (2048, 256) f32

    int*      order     = (int*)d_ws;
    TileDesc* tiles     = (TileDesc*)(order + BATCH);
    int*      num_tiles = (int*)(tiles + MAXT);

    bucket_kernel<<<dim3(1), dim3(256), 0, stream>>>(cidx, order, tiles, num_tiles);
    ctx_gemm_kernel<<<dim3(MAXT), dim3(256), 0, stream>>>(x, kern, order, tiles,
                                                          num_tiles, out);
}